// SelfAtt_38852274160189
// MI455X (gfx1250) — compile-verified
//
#include <hip/hip_runtime.h>
#include <math.h>

typedef __attribute__((ext_vector_type(2))) float v2f;
typedef __attribute__((ext_vector_type(4))) float v4f;
typedef __attribute__((ext_vector_type(8))) float v8f;

#define T_ 32
#define B_ 16
#define V_ 30000
#define D_ 300
#define DQKV_ 150
#define K_ 10
#define G_ 32
#define N_ 320            // K_*G_
#define S_ 25000
#define ROWS 512          // T_*B_
#define MPAD 304          // 19*16, padded leading dim for M and qk
#define NEG_INF (-3.4e38f)
#define LOG_EPS (-18.420680743952367f)   // log(1e-8)
#define INV_SQRT_DQKV 0.081649658092772603f

// ---------------------------------------------------------------------------
// 1) Fill output with log(EPS). 512*25000 = 12,800,000 floats = 3,200,000 float4
// ---------------------------------------------------------------------------
__global__ __launch_bounds__(256) void fill_kernel(float4* __restrict__ out) {
    const int i = blockIdx.x * 256 + threadIdx.x;   // grid sized exactly
    float4 v;
    v.x = LOG_EPS; v.y = LOG_EPS; v.z = LOG_EPS; v.w = LOG_EPS;
    out[i] = v;
}

// ---------------------------------------------------------------------------
// 2) Top-K=10 per row of predictions [512, 30000].
//    float4 non-temporal streaming (predictions are read exactly once; keep
//    the 30MB SC table resident in the 192MB L2 instead), per-thread register
//    top-10, then LDS merge with 10 block-argmax rounds.
// ---------------------------------------------------------------------------
__global__ __launch_bounds__(256) void topk_kernel(const float* __restrict__ pred,
                                                   int* __restrict__ topk_out) {
    const int row = blockIdx.x;
    const int tid = threadIdx.x;
    const v4f* p4 = (const v4f*)(pred + (size_t)row * V_);   // 7500 float4s

    float tv[K_];
    int   ti[K_];
#pragma unroll
    for (int q = 0; q < K_; ++q) { tv[q] = NEG_INF; ti[q] = 0; }

    for (int j4 = tid; j4 < (V_ / 4); j4 += 256) {
        const v4f v = __builtin_nontemporal_load(p4 + j4);
#pragma unroll
        for (int e = 0; e < 4; ++e) {
            const float val = v[e];
            if (val > tv[K_ - 1]) {
                tv[K_ - 1] = val; ti[K_ - 1] = j4 * 4 + e;
                // single bubble-up keeps descending order
#pragma unroll
                for (int q = K_ - 1; q >= 1; --q) {
                    if (tv[q] > tv[q - 1]) {
                        float fv = tv[q]; tv[q] = tv[q - 1]; tv[q - 1] = fv;
                        int   fi = ti[q]; ti[q] = ti[q - 1]; ti[q - 1] = fi;
                    }
                }
            }
        }
    }

    __shared__ float sv[256 * K_];
    __shared__ int   si[256 * K_];
    __shared__ float rv[256];
    __shared__ int   rp[256];

#pragma unroll
    for (int q = 0; q < K_; ++q) {
        sv[tid * K_ + q] = tv[q];
        si[tid * K_ + q] = ti[q];
    }
    __syncthreads();

    for (int round = 0; round < K_; ++round) {
        float bm = NEG_INF; int bp = tid * K_;
#pragma unroll
        for (int q = 0; q < K_; ++q) {
            float v = sv[tid * K_ + q];
            if (v > bm) { bm = v; bp = tid * K_ + q; }
        }
        rv[tid] = bm; rp[tid] = bp;
        __syncthreads();
        for (int off = 128; off > 0; off >>= 1) {
            if (tid < off) {
                if (rv[tid + off] > rv[tid]) { rv[tid] = rv[tid + off]; rp[tid] = rp[tid + off]; }
            }
            __syncthreads();
        }
        if (tid == 0) {
            int p0 = rp[0];
            topk_out[row * K_ + round] = si[p0];
            sv[p0] = NEG_INF;
        }
        __syncthreads();
    }
}

// ---------------------------------------------------------------------------
// WMMA layout (V_WMMA_F32_16X16X4_F32, wave32):
//   lane = l, half = l>>4, mn = l&15
//   A (16x4): a.x = A[mn][k0+2*half], a.y = A[mn][k0+2*half+1]
//   B (4x16): b.x = B[k0+2*half][mn], b.y = B[k0+2*half+1][mn]
//   D (16x16): vgpr v -> row = v + 8*half, col = mn
//
// A-row m only contributes to D-row m (and B-col n only to D-col n), so
// out-of-range lanes may compute garbage: we clamp addresses in-bounds and
// zero the padded region at store time. This keeps the WMMA loop branch-free.
// ---------------------------------------------------------------------------

// 3) M = Wq (300x150) * Wk^T (150x300)  ->  Mmat padded [304 x 304] (zeros outside)
__global__ __launch_bounds__(32) void gemm_wqwk(const float* __restrict__ Wq,
                                                const float* __restrict__ Wk,
                                                float* __restrict__ Mmat) {
    const int lane = threadIdx.x & 31;
    const int half = lane >> 4;
    const int mn   = lane & 15;
    const int row0 = blockIdx.y * 16;
    const int col0 = blockIdx.x * 16;
    const int m = row0 + mn;               // Wq row
    const int n = col0 + mn;               // Wk row (B column)
    const int mc = (m < D_) ? m : (D_ - 1);   // clamped: garbage OK, zeroed at store
    const int nc = (n < D_) ? n : (D_ - 1);
    const float* __restrict__ aptr = Wq + (size_t)mc * DQKV_;
    const float* __restrict__ bptr = Wk + (size_t)nc * DQKV_;
    const int koff = 2 * half;

    v8f acc = {0.f, 0.f, 0.f, 0.f, 0.f, 0.f, 0.f, 0.f};
    // main body: k0 = 0..144, k = k0+koff, k+1 <= 147 < 150 -> branch-free
    for (int k0 = 0; k0 < 148; k0 += 4) {
        const int k = k0 + koff;
        v2f a, b;
        a.x = aptr[k];  a.y = aptr[k + 1];
        b.x = bptr[k];  b.y = bptr[k + 1];
        acc = __builtin_amdgcn_wmma_f32_16x16x4_f32(false, a, false, b,
                                                    (short)0, acc, false, false);
    }
    // tail: k0 = 148; half==0 covers k=148,149; half==1 (k=150,151) is zero
    {
        v2f a, b;
        a.x = (half == 0) ? aptr[148] : 0.f;
        a.y = (half == 0) ? aptr[149] : 0.f;
        b.x = (half == 0) ? bptr[148] : 0.f;
        b.y = (half == 0) ? bptr[149] : 0.f;
        acc = __builtin_amdgcn_wmma_f32_16x16x4_f32(false, a, false, b,
                                                    (short)0, acc, false, false);
    }
#pragma unroll
    for (int v = 0; v < 8; ++v) {
        const int r = row0 + v + 8 * half;
        const int c = col0 + mn;
        float val = (r < D_ && c < D_) ? acc[v] : 0.f;
        Mmat[r * MPAD + c] = val;          // covers all 304x304, pads with zeros
    }
}

// 4) qk = lc (512x300) * Mmat (300x304pad) -> qk padded [512 x 304]
__global__ __launch_bounds__(32) void gemm_qk(const float* __restrict__ lc,
                                              const float* __restrict__ Mmat,
                                              float* __restrict__ qk) {
    const int lane = threadIdx.x & 31;
    const int half = lane >> 4;
    const int mn   = lane & 15;
    const int row0 = blockIdx.y * 16;      // 32 tiles -> 512 rows
    const int col0 = blockIdx.x * 16;      // 19 tiles -> 304 cols
    const int m = row0 + mn;
    const int n = col0 + mn;
    const float* __restrict__ aptr = lc + (size_t)m * D_;
    const int koff = 2 * half;

    v8f acc = {0.f, 0.f, 0.f, 0.f, 0.f, 0.f, 0.f, 0.f};
    for (int k0 = 0; k0 < D_; k0 += 4) {   // 75 unguarded WMMA iterations
        const int k = k0 + koff;
        v2f a, b;
        a.x = aptr[k];
        a.y = aptr[k + 1];
        b.x = Mmat[k * MPAD + n];
        b.y = Mmat[(k + 1) * MPAD + n];
        acc = __builtin_amdgcn_wmma_f32_16x16x4_f32(false, a, false, b,
                                                    (short)0, acc, false, false);
    }
#pragma unroll
    for (int v = 0; v < 8; ++v)
        qk[(row0 + v + 8 * half) * MPAD + col0 + mn] = acc[v];
}

// ---------------------------------------------------------------------------
// 5) Per row: neighbour gather, wave-cooperative dots vs SC (L2-resident),
//    block softmax, scatter log-probs into the pre-filled output
// ---------------------------------------------------------------------------
__global__ __launch_bounds__(256) void attn_kernel(const float* __restrict__ qk,
                                                   const int* __restrict__ topk,
                                                   const int* __restrict__ nbr,
                                                   const float* __restrict__ SC,
                                                   float* __restrict__ out) {
    const int row  = blockIdx.x;
    const int tid  = threadIdx.x;
    const int lane = tid & 31;
    const int wave = tid >> 5;             // 8 waves

    __shared__ float s_q[D_];
    __shared__ int   s_top[K_];
    __shared__ int   s_nei[N_];
    __shared__ float s_logit[N_];
    __shared__ float red[256];

    for (int d = tid; d < D_; d += 256) s_q[d] = qk[row * MPAD + d];
    if (tid < K_) s_top[tid] = topk[row * K_ + tid];
    __syncthreads();

    for (int n = tid; n < N_; n += 256) {
        const int g = s_top[n >> 5];
        s_nei[n] = nbr[(size_t)g * G_ + (n & 31)];
    }
    __syncthreads();

    // wave-cooperative dot products: 32 lanes stride over D_=300
    for (int n = wave; n < N_; n += 8) {
        const float* sc = SC + (size_t)s_nei[n] * D_;
        float sum = 0.f;
        for (int d = lane; d < D_; d += 32) sum += s_q[d] * sc[d];
#pragma unroll
        for (int m = 16; m > 0; m >>= 1) sum += __shfl_xor(sum, m, 32);
        if (lane == 0) s_logit[n] = sum * INV_SQRT_DQKV;
    }
    __syncthreads();

    // block max
    float lm = NEG_INF;
    for (int n = tid; n < N_; n += 256) lm = fmaxf(lm, s_logit[n]);
    red[tid] = lm;
    __syncthreads();
    for (int off = 128; off > 0; off >>= 1) {
        if (tid < off) red[tid] = fmaxf(red[tid], red[tid + off]);
        __syncthreads();
    }
    const float mx = red[0];
    __syncthreads();

    // block sum of exp
    float ls = 0.f;
    for (int n = tid; n < N_; n += 256) ls += __expf(s_logit[n] - mx);
    red[tid] = ls;
    __syncthreads();
    for (int off = 128; off > 0; off >>= 1) {
        if (tid < off) red[tid] += red[tid + off];
        __syncthreads();
    }
    const float lse = logf(red[0]);
    __syncthreads();

    // scatter log-probabilities (duplicates race; reference .set is also one-wins)
    for (int n = tid; n < N_; n += 256) {
        const float lp = s_logit[n] - mx - lse;
        out[(size_t)row * S_ + s_nei[n]] = lp;
    }
}

// ---------------------------------------------------------------------------
extern "C" void kernel_launch(void* const* d_in, const int* in_sizes, int n_in,
                              void* d_out, int out_size, void* d_ws, size_t ws_size,
                              hipStream_t stream) {
    const float* pred = (const float*)d_in[0];   // [32,16,30000]
    const float* lc   = (const float*)d_in[1];   // [32,16,300]
    const float* SC   = (const float*)d_in[2];   // [25000,300]
    const float* Wq   = (const float*)d_in[3];   // [300,150]
    const float* Wk   = (const float*)d_in[4];   // [300,150]
    const int*   nbr  = (const int*)d_in[5];     // [30000,32]
    float* out = (float*)d_out;                  // [512,25000]

    char* ws = (char*)d_ws;
    int*   topk = (int*)ws;                      // 512*10*4   = 20,480 B
    float* Mmat = (float*)(ws + 32768);          // 304*304*4  = 369,664 B
    float* qk   = (float*)(ws + 32768 + 369664); // 512*304*4  = 622,592 B

    // 12,800,000 floats -> 3,200,000 float4 -> 12,500 blocks of 256
    fill_kernel<<<12500, 256, 0, stream>>>((float4*)out);
    topk_kernel<<<ROWS, 256, 0, stream>>>(pred, topk);
    gemm_wqwk<<<dim3(19, 19), 32, 0, stream>>>(Wq, Wk, Mmat);
    gemm_qk<<<dim3(19, 32), 32, 0, stream>>>(lc, Mmat, qk);
    attn_kernel<<<ROWS, 256, 0, stream>>>(qk, topk, nbr, SC, out);
}